// HardcodedSmallestNN_11699490915028
// MI455X (gfx1250) — compile-verified
//
#include <hip/hip_runtime.h>

// Problem constants (fixed by the reference).
#define D_LEN   262144
#define NKEYS   512

// Tiling: 32 key-groups of 16 keys; each key-group's D range split into 32
// block slices; each 128-thread block = 4 waves, each wave owns 2048 d-elems.
#define KEY_GROUPS      32
#define NSLICES         32
#define WAVES_PER_BLOCK 4
#define BLOCK_THREADS   (WAVES_PER_BLOCK * 32)
#define S_BLOCK (D_LEN / NSLICES)            // 8192 d-elements per block
#define S_WAVE  (S_BLOCK / WAVES_PER_BLOCK)  // 2048 d-elements per wave

typedef __attribute__((ext_vector_type(2))) float v2f;
typedef __attribute__((ext_vector_type(8))) float v8f;

// --- kernel 0: zero the act accumulator workspace (ws is poisoned 0xAA) ---
__global__ void zero_acts_kernel(float* __restrict__ act) {
    if (threadIdx.x < NKEYS) act[threadIdx.x] = 0.0f;
}

// --- kernel 1: bandwidth-bound GEMV via V_WMMA_F32_16X16X4_F32 ------------
// Dot products are order-invariant over d, so we permute the WMMA K-slot
// mapping to make every load a b128:
//   per lane:  aa = float4 at  kw + row*D + d + 4*(lane>>4)
//              xx = float4 at  x +         d + 4*(lane>>4)
//   WMMA0: A regs {aa.x,aa.y}  -> K slots {d,d+1} (lanes<16) /
//                                 {d+4,d+5} (lanes>=16); B = {xx.x,xx.y}
//   WMMA1: A regs {aa.z,aa.w}  -> K slots {d+2,d+3} / {d+6,d+7};
//                                 B = {xx.z,xx.w}
// Each K value of d..d+7 appears exactly once across the two WMMAs, with A
// and B agreeing per slot, so the accumulation is the exact fp32 GEMV.
// B is column-broadcast, so every column of C holds the same partial dots:
// lane 0 owns act rows M=0..7 in c[0..7], lane 16 owns M=8..15.
__global__ __launch_bounds__(BLOCK_THREADS)
void gemv_wmma_f32_kernel(const float* __restrict__ x,
                          const float* __restrict__ kw,
                          float* __restrict__ act)
{
    const int lane  = threadIdx.x & 31;
    const int wave  = threadIdx.x >> 5;
    const int kg    = blockIdx.x & (KEY_GROUPS - 1);   // key group 0..31
    const int slice = blockIdx.x >> 5;                 // d slice   0..31
    const int d0    = slice * S_BLOCK + wave * S_WAVE;

    const int m  = lane & 15;          // key row within the 16-key group
    const int dh = (lane >> 4) << 2;   // 0 for lanes 0-15, 4 for lanes 16-31

    const float4* __restrict__ aptr =
        (const float4*)(kw + (size_t)(kg * 16 + m) * D_LEN + d0 + dh);
    const float4* __restrict__ xptr = (const float4*)(x + d0 + dh);

    v8f c0 = {};
    v8f c1 = {};
#pragma unroll 8
    for (int i = 0; i < S_WAVE / 8; ++i) {
        const float4 aa = aptr[i * 2];
        const float4 xx = xptr[i * 2];
        v2f a0; a0[0] = aa.x; a0[1] = aa.y;
        v2f a1; a1[0] = aa.z; a1[1] = aa.w;
        v2f b0; b0[0] = xx.x; b0[1] = xx.y;
        v2f b1; b1[0] = xx.z; b1[1] = xx.w;
        // (neg_a, A, neg_b, B, c_mod, C, reuse_a, reuse_b)
        c0 = __builtin_amdgcn_wmma_f32_16x16x4_f32(
                 false, a0, false, b0, (short)0, c0, false, false);
        c1 = __builtin_amdgcn_wmma_f32_16x16x4_f32(
                 false, a1, false, b1, (short)0, c1, false, false);
    }

    // Every column of C is identical; lane 0 / lane 16 publish the partials.
    if (m == 0) {
        const int mbase = (lane >> 4) * 8;   // 0 or 8
#pragma unroll
        for (int j = 0; j < 8; ++j)
            atomicAdd(&act[kg * 16 + mbase + j], c0[j] + c1[j]);
    }
}

// --- kernel 2: argmax over 512 acts + gather w1/w2 ------------------------
// Tournament with strict '>' keeps the lower index on ties, matching
// jnp.argmax's first-occurrence semantics.
__global__ void argmax_out_kernel(const float* __restrict__ act,
                                  const float* __restrict__ w1,
                                  const float* __restrict__ w2,
                                  float* __restrict__ out)
{
    __shared__ float sv[NKEYS];
    __shared__ int   si[NKEYS];
    const int t = threadIdx.x;
    sv[t] = act[t];
    si[t] = t;
    __syncthreads();
#pragma unroll
    for (int s = NKEYS / 2; s > 0; s >>= 1) {
        if (t < s) {
            if (sv[t + s] > sv[t]) { sv[t] = sv[t + s]; si[t] = si[t + s]; }
        }
        __syncthreads();
    }
    if (t == 0) {
        const int idx = si[0];
        out[0] = w1[idx];
        out[1] = w2[idx];
    }
}

extern "C" void kernel_launch(void* const* d_in, const int* in_sizes, int n_in,
                              void* d_out, int out_size, void* d_ws, size_t ws_size,
                              hipStream_t stream)
{
    const float* x  = (const float*)d_in[0];   // (1,1,262144)
    const float* kw = (const float*)d_in[1];   // (512,1,262144)
    const float* w1 = (const float*)d_in[2];   // (1,512)
    const float* w2 = (const float*)d_in[3];   // (1,512)
    float* out = (float*)d_out;                // 2 floats
    float* act = (float*)d_ws;                 // 512-float accumulator

    zero_acts_kernel<<<1, NKEYS, 0, stream>>>(act);
    gemv_wmma_f32_kernel<<<KEY_GROUPS * NSLICES, BLOCK_THREADS, 0, stream>>>(x, kw, act);
    argmax_out_kernel<<<1, NKEYS, 0, stream>>>(act, w1, w2, out);
}